// TensorDVGORes_11458972745944
// MI455X (gfx1250) — compile-verified
//
#include <hip/hip_runtime.h>
#include <stdint.h>

// Geometry from the reference: C=48, D=H=W=128, N = 2^21 points.
#define GC 48
#define GD 128
#define CH_STRIDE (GD * GD * GD)        // 2,097,152 elements per channel
#define ROW_ELEMS (GC * GD)             // 6144 floats per (z,y) tile (24 KB)

typedef unsigned int u32x4 __attribute__((ext_vector_type(4)));
typedef int          i32x4 __attribute__((ext_vector_type(4)));
typedef int          i32x8 __attribute__((ext_vector_type(8)));
typedef float        f32x4 __attribute__((ext_vector_type(4)));

// ---------------------------------------------------------------------------
// Phase 1: gridT[( (z*128+y)*128 + x )*48 + c] = k0[c,z,y,x] + former[c,z,y,x]
// One block per (z,y) row. Coalesced-enough reads (every 64B line fully
// consumed via L0), conflict-free LDS writes, single TDM store per tile.
// ---------------------------------------------------------------------------
__global__ __launch_bounds__(256)
void dvgo_sum_transpose(const float* __restrict__ k0,
                        const float* __restrict__ fm,
                        float* __restrict__ gridT)
{
    __shared__ float tile[ROW_ELEMS];          // layout: [x][c] (channel-last)
    const int row = blockIdx.x;                // row = z*128 + y
    const size_t row_base = (size_t)row * GD;  // spatial offset of (z,y,x=0)

    // f is LDS-linear: f = x*48 + c. Consecutive lanes -> consecutive LDS
    // addresses (no bank conflicts). Global reads are 4B/lane strided by 8MB
    // across lanes, but each 64B line is fully consumed out of L0 by later f.
    for (int f = threadIdx.x; f < ROW_ELEMS; f += 256) {
        int x = f / GC;
        int c = f - x * GC;
        size_t g = (size_t)c * CH_STRIDE + row_base + x;
        tile[f] = k0[g] + fm[g];
    }
    __syncthreads();

#if __has_builtin(__builtin_amdgcn_tensor_store_from_lds) && \
    __has_builtin(__builtin_amdgcn_s_wait_tensorcnt)
    // Wave 0 issues one TDM store of the contiguous 24KB tile (DMA drains the
    // tile; wave only has to hold LDS alive until TENSORcnt==0).
    if (threadIdx.x < 32u) {
        uint64_t gaddr   = (uint64_t)(uintptr_t)(gridT + (size_t)row * ROW_ELEMS);
        unsigned lds_off = (unsigned)(uintptr_t)(&tile[0]);   // LDS byte offset (addr[31:0])

        // ---- D# group 0 (128b): count=1 | lds_addr | global_addr | type=2 ----
        u32x4 g0;
        g0.x = 1u;                                   // count=1, is_restore=0, gather_mode=0
        g0.y = lds_off;                              // bits [63:32] lds_addr
        g0.z = (unsigned)(gaddr & 0xFFFFFFFFu);      // global_addr[31:0]
        g0.w = (unsigned)(gaddr >> 32) | 0x80000000u;// global_addr[56:32] | type=2 @ [127:126]

        // ---- D# group 1 (256b): 1-D tile of ROW_ELEMS 4-byte elements ----
        i32x8 g1 = {};
        const unsigned td0 = ROW_ELEMS;              // tensor_dim0 (OOB bound)
        const unsigned tl0 = ROW_ELEMS;              // tile_dim0
        g1[0] = (int)(2u << 16);                     // workgroup_mask=0, data_size=2 (4B)
        g1[1] = (int)((td0 & 0xFFFFu) << 16);        // bar_addr=0 | tensor_dim0[15:0]
        g1[2] = (int)((td0 >> 16) & 0xFFFFu);        // tensor_dim0[31:16] | tensor_dim1[15:0]=0
        g1[3] = (int)((tl0 & 0xFFFFu) << 16);        // tensor_dim1[31:16]=0 | tile_dim0
        g1[4] = 0;                                   // tile_dim1=0 (unused), tile_dim2=0
        g1[5] = (int)ROW_ELEMS;                      // tensor_dim0_stride[31:0]
        g1[6] = 0;                                   // stride0[47:32] | stride1[15:0]
        g1[7] = 0;                                   // tensor_dim1_stride[47:16]

        i32x4 z4 = {};
#if defined(__clang_major__) && (__clang_major__ >= 23)
        i32x8 z8 = {};
        __builtin_amdgcn_tensor_store_from_lds(g0, g1, z4, z4, z8, 0);
#else
        __builtin_amdgcn_tensor_store_from_lds(g0, g1, z4, z4, 0);
#endif
        __builtin_amdgcn_s_wait_tensorcnt(0);        // keep LDS alive until DMA done
    }
#else
    // Fallback: plain coalesced vector stores.
    {
        const f32x4* src = (const f32x4*)tile;
        f32x4* dst = (f32x4*)(gridT + (size_t)row * ROW_ELEMS);
        for (int i = threadIdx.x; i < ROW_ELEMS / 4; i += 256) dst[i] = src[i];
    }
#endif
}

// ---------------------------------------------------------------------------
// Phase 2: trilinear gather from channel-last grid.
// One thread per (point, 16-channel chunk): 8 corners x 4 float4 loads, all
// 64B lines fully utilized. Output stores are non-temporal so the 402MB
// output stream does not evict gather lines from the 192MB L2.
// ---------------------------------------------------------------------------
__device__ __forceinline__ f32x4 fma4(float w, f32x4 v, f32x4 a)
{
    a.x = fmaf(w, v.x, a.x);
    a.y = fmaf(w, v.y, a.y);
    a.z = fmaf(w, v.z, a.z);
    a.w = fmaf(w, v.w, a.w);
    return a;
}

__device__ __forceinline__ void trilinear_setup(const float* __restrict__ pts, int n,
                                                size_t sp[8], float wt[8])
{
    float px = pts[3 * (size_t)n + 0];
    float py = pts[3 * (size_t)n + 1];
    float pz = pts[3 * (size_t)n + 2];
    float ix = (px + 1.0f) * 0.5f * (float)(GD - 1);
    float iy = (py + 1.0f) * 0.5f * (float)(GD - 1);
    float iz = (pz + 1.0f) * 0.5f * (float)(GD - 1);
    float xf = floorf(ix), yf = floorf(iy), zf = floorf(iz);
    float wx = ix - xf, wy = iy - yf, wz = iz - zf;
    int x0 = min(max((int)xf, 0), GD - 1), x1 = min(x0 + 1, GD - 1);
    int y0 = min(max((int)yf, 0), GD - 1), y1 = min(y0 + 1, GD - 1);
    int z0 = min(max((int)zf, 0), GD - 1), z1 = min(z0 + 1, GD - 1);

    int   xs[2] = {x0, x1},          ys[2] = {y0, y1},          zs[2] = {z0, z1};
    float wxs[2] = {1.f - wx, wx},   wys[2] = {1.f - wy, wy},   wzs[2] = {1.f - wz, wz};

    int k = 0;
#pragma unroll
    for (int a = 0; a < 2; ++a)
#pragma unroll
        for (int b = 0; b < 2; ++b)
#pragma unroll
            for (int c = 0; c < 2; ++c) {
                sp[k] = ((size_t)zs[a] * GD + ys[b]) * GD + xs[c];
                wt[k] = wzs[a] * wys[b] * wxs[c];
                ++k;
            }
}

__global__ __launch_bounds__(256)
void dvgo_gather(const float* __restrict__ pts,
                 const float* __restrict__ gridT,
                 float* __restrict__ out, int N)
{
    int tid = blockIdx.x * 256 + threadIdx.x;
    if (tid >= N * 3) return;
    int n = tid / 3;
    int j = tid - n * 3;            // 16-channel chunk index (0..2)

    size_t sp[8];
    float  wt[8];
    trilinear_setup(pts, n, sp, wt);

    f32x4 a0 = {0.f, 0.f, 0.f, 0.f}, a1 = a0, a2 = a0, a3 = a0;
#pragma unroll
    for (int k = 0; k < 8; ++k) {
        const f32x4* p = (const f32x4*)(gridT + sp[k] * GC + (size_t)j * 16);
        f32x4 v0 = p[0], v1 = p[1], v2 = p[2], v3 = p[3];
        float w = wt[k];
        a0 = fma4(w, v0, a0);
        a1 = fma4(w, v1, a1);
        a2 = fma4(w, v2, a2);
        a3 = fma4(w, v3, a3);
    }

    f32x4* o = (f32x4*)(out + (size_t)n * GC + (size_t)j * 16);
    __builtin_nontemporal_store(a0, o + 0);
    __builtin_nontemporal_store(a1, o + 1);
    __builtin_nontemporal_store(a2, o + 2);
    __builtin_nontemporal_store(a3, o + 3);
}

// ---------------------------------------------------------------------------
// Fallback: fused gather straight from the two channel-major grids (used only
// if the workspace cannot hold the 402MB transposed grid).
// ---------------------------------------------------------------------------
__global__ __launch_bounds__(256)
void dvgo_fused(const float* __restrict__ pts,
                const float* __restrict__ k0,
                const float* __restrict__ fm,
                float* __restrict__ out, int N)
{
    int tid = blockIdx.x * 256 + threadIdx.x;
    if (tid >= N * 3) return;
    int n = tid / 3;
    int j = tid - n * 3;

    size_t sp[8];
    float  wt[8];
    trilinear_setup(pts, n, sp, wt);

#pragma unroll 4
    for (int c0 = 0; c0 < 16; ++c0) {
        size_t chb = (size_t)(j * 16 + c0) * CH_STRIDE;
        float acc = 0.f;
#pragma unroll
        for (int k = 0; k < 8; ++k)
            acc = fmaf(wt[k], k0[chb + sp[k]] + fm[chb + sp[k]], acc);
        __builtin_nontemporal_store(acc, out + (size_t)n * GC + j * 16 + c0);
    }
}

// ---------------------------------------------------------------------------
extern "C" void kernel_launch(void* const* d_in, const int* in_sizes, int n_in,
                              void* d_out, int out_size, void* d_ws, size_t ws_size,
                              hipStream_t stream)
{
    const float* pts = (const float*)d_in[0];   // (N,3) fp32
    const float* k0  = (const float*)d_in[1];   // (1,48,128,128,128) fp32
    const float* fm  = (const float*)d_in[2];   // (1,48,128,128,128) fp32
    float* out = (float*)d_out;                 // (N,48) fp32

    const int N = in_sizes[0] / 3;
    const int total = N * 3;
    const size_t need = (size_t)GC * GD * GD * GD * sizeof(float); // 402.6 MB

    if (ws_size >= need) {
        float* gridT = (float*)d_ws;
        dvgo_sum_transpose<<<GD * GD, 256, 0, stream>>>(k0, fm, gridT);
        dvgo_gather<<<(total + 255) / 256, 256, 0, stream>>>(pts, gridT, out, N);
    } else {
        dvgo_fused<<<(total + 255) / 256, 256, 0, stream>>>(pts, k0, fm, out, N);
    }
}